// ConvLayer_32220844654996
// MI455X (gfx1250) — compile-verified
//
#include <hip/hip_runtime.h>

typedef __attribute__((ext_vector_type(2))) float v2f;
typedef __attribute__((ext_vector_type(8))) float v8f;

#define NNODES 100000
#define NEDGES 1600000
#define D 64
#define LDS_STRIDE 68   // 64 + 4 pad: bank = (4m + k) % 64 -> conflict-free frag reads

// ---------------------------------------------------------------------------
// Kernel 0: zero the scatter accumulators (summed[N*64] and deg[N]) in d_ws.
// ---------------------------------------------------------------------------
__global__ void sage_zero_kernel(float4* __restrict__ ws, int n4) {
    int i = blockIdx.x * blockDim.x + threadIdx.x;
    if (i < n4) ws[i] = make_float4(0.f, 0.f, 0.f, 0.f);
}

// ---------------------------------------------------------------------------
// Kernel 1: edge-parallel scatter-add (copy_src -> sum over dst).
// One thread = (edge, 4-float feature chunk). 16 threads cover an edge's 64
// features; their float4 loads of h_neigh[src] form one contiguous 256B row.
// summed (25.6 MB) is L2-resident on MI455X (192 MB L2), so the f32 atomics
// resolve at L2 bandwidth.
// ---------------------------------------------------------------------------
__global__ void sage_scatter_kernel(const float* __restrict__ h_neigh,
                                    const long long* __restrict__ src,
                                    const long long* __restrict__ dst,
                                    float* __restrict__ summed,
                                    float* __restrict__ deg) {
    long long tid  = (long long)blockIdx.x * blockDim.x + threadIdx.x;
    long long edge = tid >> 4;
    if (edge >= NEDGES) return;
    int f = (int)(tid & 15) * 4;

    long long s = src[edge];
    long long d = dst[edge];

    const float4 v = *(const float4*)(h_neigh + s * D + f);
    float* o = summed + d * D + f;
    atomicAdd(o + 0, v.x);
    atomicAdd(o + 1, v.y);
    atomicAdd(o + 2, v.z);
    atomicAdd(o + 3, v.w);
    if (f == 0) atomicAdd(deg + d, 1.0f);
}

// ---------------------------------------------------------------------------
// Kernel 2: fused  z = relu(h_self @ W_self^T + (summed/deg) @ W_neigh^T),
// then row-wise L2 normalize.  One block = 16 nodes, 4 waves; wave w computes
// output columns [16w, 16w+16) with V_WMMA_F32_16X16X4_F32, K-chained over
// the self (K=64) and neighbor (K=64) contractions into one f32 accumulator.
//
// Frag layouts (ISA 7.12.2, wave32):
//   A 16x4 f32 : lane&15 = M,  K = (lane>>4)*2 + vgpr
//   B 4x16 f32 : lane&15 = N,  K = (lane>>4)*2 + vgpr
//   C/D 16x16  : vgpr i -> M = i + (lane>>4)*8, N = lane&15
// ---------------------------------------------------------------------------
__global__ __launch_bounds__(128)
void sage_gemm_kernel(const float* __restrict__ h_self,
                      const float* __restrict__ summed,
                      const float* __restrict__ deg,
                      const float* __restrict__ W_self,
                      const float* __restrict__ W_neigh,
                      float* __restrict__ out) {
    __shared__ float sSelf[16 * LDS_STRIDE];
    __shared__ float sAgg [16 * LDS_STRIDE];
    __shared__ float sInv [16];

    const int t        = threadIdx.x;
    const int row_base = blockIdx.x * 16;

    // ---- Stage tiles into LDS with coalesced global loads -----------------
    {
        const int r  = t >> 3;           // 0..15
        const int cb = (t & 7) * 8;      // 0,8,..,56
        const long long g = (long long)(row_base + r) * D + cb;

        float4 a0 = *(const float4*)(h_self + g);
        float4 a1 = *(const float4*)(h_self + g + 4);
        *(float4*)(&sSelf[r * LDS_STRIDE + cb])     = a0;
        *(float4*)(&sSelf[r * LDS_STRIDE + cb + 4]) = a1;

        const float inv = 1.0f / fmaxf(deg[row_base + r], 1.0f);
        float4 b0 = *(const float4*)(summed + g);
        float4 b1 = *(const float4*)(summed + g + 4);
        b0.x *= inv; b0.y *= inv; b0.z *= inv; b0.w *= inv;
        b1.x *= inv; b1.y *= inv; b1.z *= inv; b1.w *= inv;
        *(float4*)(&sAgg[r * LDS_STRIDE + cb])     = b0;
        *(float4*)(&sAgg[r * LDS_STRIDE + cb + 4]) = b1;
    }
    __syncthreads();

    const int lane     = t & 31;
    const int wid      = t >> 5;
    const int col_base = wid * 16;
    const int mn       = lane & 15;        // M (A frag) / N (B frag)
    const int koff     = (lane >> 4) * 2;  // K sub-offset per lane half

    v8f c = {0.f, 0.f, 0.f, 0.f, 0.f, 0.f, 0.f, 0.f};

    const float* As = &sSelf[mn * LDS_STRIDE];
    const float* An = &sAgg [mn * LDS_STRIDE];
    const float* Bs = W_self  + (long long)(col_base + mn) * D;  // B[k][n] = W[n][k]
    const float* Bn = W_neigh + (long long)(col_base + mn) * D;

#pragma unroll
    for (int k0 = 0; k0 < D; k0 += 4) {
        v2f a, b;
        a.x = As[k0 + koff]; a.y = As[k0 + koff + 1];
        b.x = Bs[k0 + koff]; b.y = Bs[k0 + koff + 1];
        c = __builtin_amdgcn_wmma_f32_16x16x4_f32(false, a, false, b,
                                                  (short)0, c, false, false);
    }
#pragma unroll
    for (int k0 = 0; k0 < D; k0 += 4) {
        v2f a, b;
        a.x = An[k0 + koff]; a.y = An[k0 + koff + 1];
        b.x = Bn[k0 + koff]; b.y = Bn[k0 + koff + 1];
        c = __builtin_amdgcn_wmma_f32_16x16x4_f32(false, a, false, b,
                                                  (short)0, c, false, false);
    }

    __syncthreads();              // tile reads done; reuse sSelf as z tile
    float* zT = sSelf;

    const int mbase = (lane >> 4) * 8;
#pragma unroll
    for (int i = 0; i < 8; ++i) {
        float v = fmaxf(c[i], 0.0f);                       // ReLU
        zT[(mbase + i) * LDS_STRIDE + col_base + mn] = v;
    }
    __syncthreads();

    // ---- Row L2 norms (16 rows, one thread each) --------------------------
    if (t < 16) {
        float ss = 0.f;
#pragma unroll
        for (int n = 0; n < D; ++n) {
            float v = zT[t * LDS_STRIDE + n];
            ss += v * v;
        }
        sInv[t] = (ss > 0.f) ? rsqrtf(ss) : 1.0f;          // zero-norm guard
    }
    __syncthreads();

    // ---- Normalized coalesced store ---------------------------------------
    {
        const int r  = t >> 3;
        const int cb = (t & 7) * 8;
        const float inv = sInv[r];
        float* o = out + (long long)(row_base + r) * D + cb;
#pragma unroll
        for (int j = 0; j < 8; ++j)
            o[j] = zT[r * LDS_STRIDE + cb + j] * inv;
    }
}

// ---------------------------------------------------------------------------
extern "C" void kernel_launch(void* const* d_in, const int* in_sizes, int n_in,
                              void* d_out, int out_size, void* d_ws, size_t ws_size,
                              hipStream_t stream) {
    const float*     h_neigh = (const float*)    d_in[0];
    const float*     h_self  = (const float*)    d_in[1];
    const long long* src     = (const long long*)d_in[2];
    const long long* dst     = (const long long*)d_in[3];
    const float*     W_neigh = (const float*)    d_in[4];
    const float*     W_self  = (const float*)    d_in[5];
    float*           out     = (float*)d_out;

    float* summed = (float*)d_ws;                         // [N * 64]
    float* deg    = summed + (size_t)NNODES * D;          // [N]

    // Zero accumulators (required every call; harness doesn't re-poison).
    const int n4 = (NNODES * D + NNODES) / 4;
    sage_zero_kernel<<<(n4 + 255) / 256, 256, 0, stream>>>((float4*)d_ws, n4);

    // Edge scatter-add: one thread per (edge, 4-feature chunk).
    const long long work = (long long)NEDGES * 16;
    sage_scatter_kernel<<<(int)((work + 255) / 256), 256, 0, stream>>>(
        h_neigh, src, dst, summed, deg);

    // Fused WMMA GEMM + ReLU + L2 normalize: 16-node tiles, 4 waves/block.
    sage_gemm_kernel<<<NNODES / 16, 128, 0, stream>>>(
        h_self, summed, deg, W_self, W_neigh, out);
}